// MPNN_68143951119068
// MI455X (gfx1250) — compile-verified
//
#include <hip/hip_runtime.h>
#include <math.h>

#define NAGENT 10
#define PTOT   65536
#define INFEAT 16
#define EDIM   64
#define HDIM   128
#define PB     8              // positions per block
#define RB     (NAGENT * PB)  // 80 rows per block
#define NT     (RB / 16)      // 5 row tiles of 16
#define HS     68             // padded LDS row stride (floats)

typedef __attribute__((ext_vector_type(2))) float v2f;
typedef __attribute__((ext_vector_type(8))) float v8f;

// D = A(16x4, f32) x B(4x16, f32) + C(16x16, f32), full-precision path on CDNA5.
__device__ __forceinline__ v8f wmma_f32_k4(v2f a, v2f b, v8f c) {
  return __builtin_amdgcn_wmma_f32_16x16x4_f32(false, a, false, b, (short)0, c,
                                               false, false);
}

// Output row pointer: leader rows at [0, P), followers at P + (a-1)*P + p.
__device__ __forceinline__ float* out_row(float* out, int a, int p) {
  if (a == 0) return out + (size_t)p * HDIM;
  return out + (size_t)PTOT * HDIM + ((size_t)(a - 1) * PTOT + (size_t)p) * HDIM;
}

// One wave computes a 16-column slab (j0..j0+15) of D for its row tiles.
// B slab (K x 16 of the weight) is held in VGPRs across the whole sweep.
// Rows of W beyond kmax are treated as zero (used for the 14->16 K padding).
// j0/rh must be wave-uniform scalars so the tile loop needs no exec masking
// (WMMA requires EXEC == all-1s).
template <int KS, typename AF, typename ST>
__device__ __forceinline__ void run_gemm(const float* __restrict__ Wg, int kmax,
                                         const float* __restrict__ bias,
                                         int j0, int rh, int l16, int lhalf,
                                         AF afetch, ST store) {
  v2f Bs[KS];
#pragma unroll
  for (int kk = 0; kk < KS; ++kk) {
    int kb  = kk * 4 + lhalf * 2;
    int col = j0 + l16;
    float b0 = (kb + 0 < kmax) ? Wg[(size_t)(kb + 0) * EDIM + col] : 0.0f;
    float b1 = (kb + 1 < kmax) ? Wg[(size_t)(kb + 1) * EDIM + col] : 0.0f;
    Bs[kk].x = b0;
    Bs[kk].y = b1;
  }
  float binit = bias ? bias[j0 + l16] : 0.0f;
  for (int t = rh; t < NT; t += 2) {
    v8f c;
#pragma unroll
    for (int r = 0; r < 8; ++r) c[r] = binit;
#pragma unroll
    for (int kk = 0; kk < KS; ++kk) {
      v2f a = afetch(t, kk);
      c = wmma_f32_k4(a, Bs[kk], c);
    }
    store(t, c);
  }
}

__global__ __launch_bounds__(256) void mpnn_wmma_kernel(
    const float* __restrict__ inp, const float* __restrict__ W_enc,
    const float* __restrict__ b_enc, const float* __restrict__ W_goal,
    const float* __restrict__ b_goal, const float* __restrict__ Wq,
    const float* __restrict__ Wk, const float* __restrict__ Wv,
    const float* __restrict__ Wo, const float* __restrict__ W_upd,
    const float* __restrict__ b_upd, float* __restrict__ out) {
  // LDS: ~90 KB total -> multiple workgroups can coreside on a 320 KB WGP.
  __shared__ __align__(16) float xls[RB * INFEAT];  // staged inputs; reused as attn scratch
  __shared__ __align__(16) float hls[RB * HS];      // h
  __shared__ __align__(16) float Qls[RB * HS];      // Q, then m_att
  __shared__ __align__(16) float Kls[RB * HS];      // K, then m
  __shared__ __align__(16) float Vls[RB * HS];      // V

  const int tid   = threadIdx.x;
  const int lane  = tid & 31;
  const int l16   = lane & 15;
  const int lhalf = lane >> 4;
  // Force wave id (hence j0/rh) into SGPRs: the row-tile loop then compiles
  // to a uniform scalar loop, keeping EXEC all-1s around every WMMA.
  const int wave  = __builtin_amdgcn_readfirstlane(tid >> 5);
  const int j0    = (wave & 3) * 16;  // column tile of this wave
  const int rh    = wave >> 2;        // row-tile phase (tiles rh, rh+2, ...)
  const int p0    = blockIdx.x * PB;

  // ---- Stage x (rows = a*PB + lp); zero cols 14,15 (goal cols not in encoder).
  for (int i = tid; i < RB * 4; i += 256) {
    int r = i >> 2, c4 = i & 3;
    int a = r >> 3, lp = r & 7;
    float4 v = *(const float4*)(inp + ((size_t)a * PTOT + p0 + lp) * INFEAT + c4 * 4);
    if (c4 == 3) { v.z = 0.0f; v.w = 0.0f; }
    *(float4*)(xls + r * INFEAT + c4 * 4) = v;
  }
  __syncthreads();

  // A-fragment fetchers / C-store helpers (built per LDS buffer, no pointer
  // arrays of __shared__ objects -- those break the device linker).
  auto afetch_from = [&](const float* src, int stride) {
    return [=](int t, int kk) {
      v2f a;
      int k = kk * 4 + lhalf * 2;
      const float* p = src + (t * 16 + l16) * stride + k;
      a.x = p[0];
      a.y = p[1];
      return a;
    };
  };
  auto store_to = [&](float* dst, bool relu) {
    return [=](int t, v8f c) {
#pragma unroll
      for (int r = 0; r < 8; ++r) {
        float v = c[r];
        if (relu) v = fmaxf(v, 0.0f);
        dst[(t * 16 + r + lhalf * 8) * HS + j0 + l16] = v;
      }
    };
  };

  // ---- Encoder: h = relu(x @ W_enc + b_enc), K = 16 (rows 14,15 zero).
  run_gemm<4>(W_enc, INFEAT - 2, b_enc, j0, rh, l16, lhalf,
              afetch_from(xls, INFEAT), store_to(hls, true));
  __syncthreads();

  // ---- Q/K/V: three 64x64 GEMMs sharing the h operand (manually unrolled).
  run_gemm<16>(Wq, EDIM, nullptr, j0, rh, l16, lhalf,
               afetch_from(hls, HS), store_to(Qls, false));
  run_gemm<16>(Wk, EDIM, nullptr, j0, rh, l16, lhalf,
               afetch_from(hls, HS), store_to(Kls, false));
  run_gemm<16>(Wv, EDIM, nullptr, j0, rh, l16, lhalf,
               afetch_from(hls, HS), store_to(Vls, false));
  __syncthreads();

  // ---- Attention (tiny 10x10 per position): VALU path.
  float* scr = xls;  // 800 floats of scratch, x is dead now
  for (int i = tid; i < PB * NAGENT * NAGENT; i += 256) {
    int lp = i / (NAGENT * NAGENT);
    int q  = (i / NAGENT) % NAGENT;
    int k  = i % NAGENT;
    float s;
    if (k == 0 || k == q) {
      s = -1e30f;  // mask: diag | key==leader
    } else {
      const float* qp = Qls + (q * PB + lp) * HS;
      const float* kp = Kls + (k * PB + lp) * HS;
      float acc = 0.0f;
#pragma unroll 8
      for (int c = 0; c < EDIM; ++c) acc += qp[c] * kp[c];
      s = acc * 0.125f;  // 1/sqrt(64)
    }
    scr[i] = s;
  }
  __syncthreads();
  if (tid < PB * NAGENT) {  // softmax over k, one thread per (lp, q)
    float* row = scr + tid * NAGENT;
    float mx = -1e30f;
    for (int k = 0; k < NAGENT; ++k) mx = fmaxf(mx, row[k]);
    float sum = 0.0f;
    for (int k = 0; k < NAGENT; ++k) {
      float e = __expf(row[k] - mx);
      row[k] = e;
      sum += e;
    }
    float inv = 1.0f / sum;
    for (int k = 0; k < NAGENT; ++k) row[k] *= inv;
  }
  __syncthreads();
  // m_att = attn @ V  (overwrites Q's LDS region; Q is dead).
  for (int i = tid; i < PB * NAGENT * EDIM; i += 256) {
    int c  = i % EDIM;
    int q  = (i / EDIM) % NAGENT;
    int lp = i / (EDIM * NAGENT);
    const float* arow = scr + (lp * NAGENT + q) * NAGENT;
    float acc = 0.0f;
#pragma unroll
    for (int k = 0; k < NAGENT; ++k)
      acc += arow[k] * Vls[(k * PB + lp) * HS + c];
    Qls[(q * PB + lp) * HS + c] = acc;
  }
  __syncthreads();

  // ---- m = m_att @ Wo  (into K's LDS region; K is dead).
  run_gemm<16>(Wo, EDIM, nullptr, j0, rh, l16, lhalf,
               afetch_from(Qls, HS), store_to(Kls, false));
  __syncthreads();

  // ---- h1 = relu([h | m] @ W_upd + b_upd), K = 128; store straight to global.
  run_gemm<32>(W_upd, 2 * EDIM, b_upd, j0, rh, l16, lhalf,
      [&](int t, int kk) {
        v2f a;
        int k = kk * 4 + lhalf * 2;
        const float* p = (k < EDIM) ? (hls + (t * 16 + l16) * HS + k)
                                    : (Kls + (t * 16 + l16) * HS + (k - EDIM));
        a.x = p[0];
        a.y = p[1];
        return a;
      },
      [&](int t, v8f c) {
#pragma unroll
        for (int r = 0; r < 8; ++r) {
          int row = t * 16 + r + lhalf * 8;
          int a = row >> 3, lp = row & 7;
          out_row(out, a, p0 + lp)[j0 + l16] = fmaxf(c[r], 0.0f);
        }
      });

  // ---- h2 for followers (cols 64:128) = agent-0's h at the same position.
  for (int i = tid; i < 9 * PB * 16; i += 256) {
    int c4 = i & 15;
    int lp = (i >> 4) & 7;
    int a  = (i >> 7) + 1;
    float4 v = *(const float4*)(hls + lp * HS + c4 * 4);  // agent 0 row = lp
    *(float4*)(out_row(out, a, p0 + lp) + EDIM + c4 * 4) = v;
  }
  // ---- h2 for leader (cols 64:128) = relu(goal @ W_goal + b_goal), K = 2.
  for (int i = tid; i < PB * EDIM; i += 256) {
    int c  = i & 63;
    int lp = i >> 6;
    float g0 = inp[(size_t)(p0 + lp) * INFEAT + 14];
    float g1 = inp[(size_t)(p0 + lp) * INFEAT + 15];
    float s  = fmaxf(g0 * W_goal[c] + g1 * W_goal[EDIM + c] + b_goal[c], 0.0f);
    out[(size_t)(p0 + lp) * HDIM + EDIM + c] = s;
  }
}

extern "C" void kernel_launch(void* const* d_in, const int* in_sizes, int n_in,
                              void* d_out, int out_size, void* d_ws,
                              size_t ws_size, hipStream_t stream) {
  const float* inp    = (const float*)d_in[0];
  const float* W_enc  = (const float*)d_in[1];
  const float* b_enc  = (const float*)d_in[2];
  const float* W_goal = (const float*)d_in[3];
  const float* b_goal = (const float*)d_in[4];
  const float* Wq     = (const float*)d_in[5];
  const float* Wk     = (const float*)d_in[6];
  const float* Wv     = (const float*)d_in[7];
  const float* Wo     = (const float*)d_in[8];
  const float* W_upd  = (const float*)d_in[9];
  const float* b_upd  = (const float*)d_in[10];
  float* out = (float*)d_out;

  dim3 grid(PTOT / PB);
  dim3 block(256);
  hipLaunchKernelGGL(mpnn_wmma_kernel, grid, block, 0, stream, inp, W_enc,
                     b_enc, W_goal, b_goal, Wq, Wk, Wv, Wo, W_upd, b_upd, out);
}